// ELM_AE_Spatial_EVOLVED_18184891531769
// MI455X (gfx1250) — compile-verified
//
#include <hip/hip_runtime.h>
#include <math.h>

// ELM-AE spatial fit for MI455X (gfx1250, wave32, WMMA).
// x: (32,1024,32,32) f32, W: (1025,12) f32 -> beta: (32,1,12288) f32.
// Memory-bound (128MB input, ~12 flop/byte) -> fp32 WMMA 16x16x4 keeps full
// precision while running the two 1024x1024x16 GEMMs on the matrix pipe.
// Global->LDS staging uses gfx1250 async-to-LDS DMA, double-buffered so the
// copy of tile t+1 overlaps the WMMA chain of tile t.

#define BATCH 32
#define NCH   1024   // channels (K of GEMM1, M of GEMM2)
#define NPIX  1024   // pixels   (M of GEMM1, K of GEMM2)
#define Q     12
#define QP    16     // Q padded to WMMA N=16 (cols 12-14 zero, col 15 == 1.0)
#define RIDGE 1e-6f

typedef __attribute__((ext_vector_type(2))) float v2f;
typedef __attribute__((ext_vector_type(8))) float v8f;
typedef __attribute__((ext_vector_type(4))) int   v4i;

// ---- CDNA5 async global->LDS path (guarded; falls back to float4 copies) ----
#if defined(__has_builtin)
#  if __has_builtin(__builtin_amdgcn_global_load_async_to_lds_b128)
#    define HAVE_ASYNC_LDS 1
#  endif
#  if __has_builtin(__builtin_amdgcn_s_wait_asynccnt)
#    define HAVE_WAIT_ASYNC 1
#  endif
#endif

__device__ __forceinline__ void stage16(const float* g, float* l) {
#ifdef HAVE_ASYNC_LDS
  __builtin_amdgcn_global_load_async_to_lds_b128((v4i*)g, (v4i*)l, 0, 0);
#else
  *(float4*)l = *(const float4*)g;
#endif
}
__device__ __forceinline__ void async_wait0() {
#ifdef HAVE_ASYNC_LDS
#  ifdef HAVE_WAIT_ASYNC
  __builtin_amdgcn_s_wait_asynccnt(0);
#  else
  asm volatile("s_wait_asynccnt 0" ::: "memory");
#  endif
#endif
}

__device__ __forceinline__ v8f wmma_f32_k4(v2f a, v2f b, v8f c) {
  // D(16x16,f32) += A(16x4,f32) * B(4x16,f32) + C
  return __builtin_amdgcn_wmma_f32_16x16x4_f32(
      /*neg_a=*/false, a, /*neg_b=*/false, b,
      /*c_mod=*/(short)0, c, /*reuse_a=*/false, /*reuse_b=*/false);
}

// ---------------- K1: per-channel mean / 1/std (ddof=1) ----------------
__global__ void k_stats(const float* __restrict__ x,
                        float* __restrict__ mu, float* __restrict__ isd) {
  int b = blockIdx.y;
  int wave = threadIdx.x >> 5, lane = threadIdx.x & 31;
  int ch = blockIdx.x * 8 + wave;
  const float* p = x + ((size_t)b * NCH + ch) * NPIX;
  float s = 0.f, s2 = 0.f;
#pragma unroll
  for (int i = 0; i < 8; ++i) {
    float4 v = ((const float4*)p)[lane + i * 32];
    s  += v.x + v.y + v.z + v.w;
    s2 += v.x*v.x + v.y*v.y + v.z*v.z + v.w*v.w;
  }
#pragma unroll
  for (int off = 16; off; off >>= 1) {
    s  += __shfl_xor(s,  off, 32);
    s2 += __shfl_xor(s2, off, 32);
  }
  if (lane == 0) {
    float m   = s * (1.f / NPIX);
    float var = (s2 - (float)NPIX * m * m) * (1.f / (NPIX - 1));
    float sd  = sqrtf(fmaxf(var, 0.f));
    float inv = (sd > 0.f) ? 1.f / sd : 0.f;   // nan_to_num behavior
    if (!(inv < 3.4e38f)) inv = 0.f;
    mu [b * NCH + ch] = m;
    isd[b * NCH + ch] = inv;
  }
}

// ---------------- K2a: W'[c,q] = W[c,q] * isd[c], padded to 16 ----------------
__global__ void k_prep_w(const float* __restrict__ W,
                         const float* __restrict__ isd, float* __restrict__ Wp) {
  int b = blockIdx.y;
  int c = blockIdx.x * 128 + threadIdx.x;
  float s = isd[b * NCH + c];
  float* dst = Wp + (size_t)(b * NCH + c) * QP;
  const float* src = W + c * Q;
#pragma unroll
  for (int q = 0; q < Q; ++q) dst[q] = src[q] * s;
#pragma unroll
  for (int q = Q; q < QP; ++q) dst[q] = 0.f;
}

// ---------------- K2b: bias'[q] = W[N,q] - sum_c mu[c]*W'[c,q] ----------------
__global__ void k_prep_bias(const float* __restrict__ W, const float* __restrict__ mu,
                            const float* __restrict__ Wp, float* __restrict__ biasp) {
  int b = blockIdx.x, tid = threadIdx.x;
  float acc[Q];
#pragma unroll
  for (int q = 0; q < Q; ++q) acc[q] = 0.f;
  for (int c = tid; c < NCH; c += 256) {
    float m = mu[b * NCH + c];
    const float* wp = Wp + (size_t)(b * NCH + c) * QP;
#pragma unroll
    for (int q = 0; q < Q; ++q) acc[q] += m * wp[q];
  }
#pragma unroll
  for (int q = 0; q < Q; ++q)
#pragma unroll
    for (int off = 16; off; off >>= 1) acc[q] += __shfl_xor(acc[q], off, 32);
  __shared__ float sred[Q * 8];
  int lane = tid & 31, wave = tid >> 5;
  if (lane == 0)
#pragma unroll
    for (int q = 0; q < Q; ++q) sred[q * 8 + wave] = acc[q];
  __syncthreads();
  if (tid < QP) {
    float v = 0.f;
    if (tid < Q) {
      v = W[NCH * Q + tid];
      for (int w = 0; w < 8; ++w) v -= sred[tid * 8 + w];
    }
    biasp[b * QP + tid] = v;
  }
}

// ---------------- K3: H = sigmoid(x^T @ W' + bias'), padded ----------------
// D tile 16p x 16q; A[m=p,k=c]=x[b,c,p] (LDS), B[k=c,n=q]=W' (LDS).
// Double-buffered async staging; per-tile: preload 8 fragment pairs, then
// an uninterrupted 8-deep WMMA accumulate chain.
#define K3_KT  32
#define K3_LDA 72   // row pad: 2*72 % 64 = 16 -> half-waves hit disjoint banks

__global__ void k_gemm_h(const float* __restrict__ x, const float* __restrict__ Wp,
                         const float* __restrict__ biasp, float* __restrict__ H) {
  __shared__ float ldsA[2][K3_KT * K3_LDA];
  __shared__ float ldsB[2][K3_KT * QP];
  int b = blockIdx.y;
  int pbase = blockIdx.x * 64;
  int tid = threadIdx.x, lane = tid & 31, wave = tid >> 5;
  const float* xb  = x  + (size_t)b * NCH * NPIX;
  const float* wpb = Wp + (size_t)b * NCH * QP;
  int n    = lane & 15;
  int koff = (lane >> 4) * 2;
  int acol = wave * 16 + n;

  // staging indices (128 threads)
  int ar = tid >> 4, ac4 = tid & 15;     // x tile: +4 row-passes of 8 rows
  int br = tid >> 2, bc4 = tid & 3;      // W' tile: 32 rows x 4 quads

  // prologue: stage tile 0 into buffer 0
#pragma unroll
  for (int it = 0; it < 4; ++it)
    stage16(xb + (size_t)(ar + it * 8) * NPIX + pbase + ac4 * 4,
            &ldsA[0][(ar + it * 8) * K3_LDA + ac4 * 4]);
  stage16(wpb + (size_t)br * QP + bc4 * 4, &ldsB[0][br * QP + bc4 * 4]);
  async_wait0();
  __syncthreads();

  v8f acc = {};
  int buf = 0;
  for (int k0 = 0; k0 < NCH; k0 += K3_KT) {
    int nxt = buf ^ 1;
    if (k0 + K3_KT < NCH) {            // overlap: DMA tile t+1 during compute
      int kn = k0 + K3_KT;
#pragma unroll
      for (int it = 0; it < 4; ++it)
        stage16(xb + (size_t)(kn + ar + it * 8) * NPIX + pbase + ac4 * 4,
                &ldsA[nxt][(ar + it * 8) * K3_LDA + ac4 * 4]);
      stage16(wpb + (size_t)(kn + br) * QP + bc4 * 4,
              &ldsB[nxt][br * QP + bc4 * 4]);
    }
    const float* Ab = &ldsA[buf][koff * K3_LDA + acol];
    const float* Bb = &ldsB[buf][koff * QP + n];
    v2f av[8], bv[8];
#pragma unroll
    for (int j = 0; j < 8; ++j) {      // batched fragment loads (imm offsets)
      av[j].x = Ab[(4 * j)     * K3_LDA];
      av[j].y = Ab[(4 * j + 1) * K3_LDA];
      bv[j].x = Bb[(4 * j)     * QP];
      bv[j].y = Bb[(4 * j + 1) * QP];
    }
#pragma unroll
    for (int j = 0; j < 8; ++j) acc = wmma_f32_k4(av[j], bv[j], acc);
    async_wait0();
    __syncthreads();
    buf = nxt;
  }

  float bias = biasp[b * QP + n];
  int prow0 = pbase + wave * 16 + ((lane >> 4) * 8);
#pragma unroll
  for (int r = 0; r < 8; ++r) {
    float v;
    if (n < Q) {
      float t = acc[r] + bias;
      v = 1.f / (1.f + __expf(-t));
    } else {
      v = (n == 15) ? 1.f : 0.f;       // ones column -> Gram row 15 == s[q]
    }
    H[((size_t)b * NPIX + prow0 + r) * QP + n] = v;
  }
}

// ---------------- K4: G = Hpad^T Hpad (A frag == B frag) ----------------
__global__ void k_gram(const float* __restrict__ H, float* __restrict__ G) {
  int b = blockIdx.x;
  int lane = threadIdx.x & 31;
  int n = lane & 15, koff = (lane >> 4) * 2;
  const float* Hb = H + (size_t)b * NPIX * QP;
  v8f acc = {};
  for (int k = 0; k < NPIX; k += 32) {
    const float* base = Hb + (size_t)(k + koff) * QP + n;
    v2f f[8];
#pragma unroll
    for (int j = 0; j < 8; ++j) {
      f[j].x = base[(4 * j)     * QP];
      f[j].y = base[(4 * j + 1) * QP];
    }
#pragma unroll
    for (int j = 0; j < 8; ++j) acc = wmma_f32_k4(f[j], f[j], acc);
  }
  int m0 = (lane >> 4) * 8;
#pragma unroll
  for (int r = 0; r < 8; ++r)
    G[b * 256 + (m0 + r) * 16 + n] = acc[r];
}

// ---------------- K5: 12x12 Gauss-Jordan inverse of (G + ridge*I) ----------------
__global__ void k_solve(const float* __restrict__ G, float* __restrict__ Ginv) {
  __shared__ float a[Q][25];
  __shared__ float pivinv;
  int b = blockIdx.x, tid = threadIdx.x;
  int i = tid >> 5, j = tid & 31;
  bool act = (j < 24);
  if (act) {
    float v;
    if (j < Q) v = G[b * 256 + i * 16 + j] + ((i == j) ? RIDGE : 0.f);
    else       v = ((j - Q) == i) ? 1.f : 0.f;
    a[i][j] = v;
  }
#pragma unroll 1
  for (int k = 0; k < Q; ++k) {
    __syncthreads();
    if (tid == 0) pivinv = 1.f / a[k][k];
    __syncthreads();
    if (i == k && act) a[k][j] *= pivinv;
    __syncthreads();
    float f = (act && i != k) ? a[i][k] : 0.f;
    __syncthreads();
    if (act && i != k) a[i][j] -= f * a[k][j];
  }
  __syncthreads();
  if (act && j >= Q) Ginv[b * 256 + i * 16 + (j - Q)] = a[i][j];
}

// ---------------- K6: M2[c,q] = sum_p x[b,c,p] * Hpad[p,q] ----------------
// A[m=c,k=p]=x (contiguous in k -> v2f global loads with imm offsets),
// B[k=p,n=q]=H double-buffered in LDS via async DMA.
#define K6_KT 32
__global__ void k_gemm_m2(const float* __restrict__ x, const float* __restrict__ H,
                          float* __restrict__ M2) {
  __shared__ float Hs[2][K6_KT * QP];
  int b = blockIdx.y;
  int cbase = blockIdx.x * 64;
  int tid = threadIdx.x, lane = tid & 31, wave = tid >> 5;
  int n = lane & 15, koff = (lane >> 4) * 2;
  const float* xb = x + (size_t)b * NCH * NPIX;
  const float* Hb = H + (size_t)b * NPIX * QP;
  const float* arow = xb + (size_t)(cbase + wave * 16 + n) * NPIX;
  int br = tid >> 2, bc4 = tid & 3;      // H tile: 32 rows x 4 quads

  stage16(Hb + (size_t)br * QP + bc4 * 4, &Hs[0][br * QP + bc4 * 4]);
  async_wait0();
  __syncthreads();

  v8f acc = {};
  int buf = 0;
  for (int k0 = 0; k0 < NPIX; k0 += K6_KT) {
    int nxt = buf ^ 1;
    if (k0 + K6_KT < NPIX)
      stage16(Hb + (size_t)(k0 + K6_KT + br) * QP + bc4 * 4,
              &Hs[nxt][br * QP + bc4 * 4]);
    const float* Ab = arow + k0 + koff;
    const float* Bb = &Hs[buf][koff * QP + n];
    v2f av[8], bv[8];
#pragma unroll
    for (int j = 0; j < 8; ++j) {
      av[j]   = *(const v2f*)(Ab + 4 * j);
      bv[j].x = Bb[(4 * j)     * QP];
      bv[j].y = Bb[(4 * j + 1) * QP];
    }
#pragma unroll
    for (int j = 0; j < 8; ++j) acc = wmma_f32_k4(av[j], bv[j], acc);
    async_wait0();
    __syncthreads();
    buf = nxt;
  }
  int c0 = cbase + wave * 16 + ((lane >> 4) * 8);
#pragma unroll
  for (int r = 0; r < 8; ++r)
    M2[(size_t)(b * NCH + c0 + r) * QP + n] = acc[r];
}

// -------- K7: beta[q,c] = sum_r Ginv[q,r] * isd[c]*(M2[c,r]-mu[c]*s[r]) --------
__global__ void k_finalize(const float* __restrict__ M2, const float* __restrict__ Ginv,
                           const float* __restrict__ G, const float* __restrict__ mu,
                           const float* __restrict__ isd, float* __restrict__ out) {
  __shared__ float Gi[Q * QP];
  __shared__ float ss[Q];
  int b = blockIdx.y, tid = threadIdx.x;
  if (tid < Q * QP) Gi[tid] = Ginv[b * 256 + tid];
  if (tid < Q)      ss[tid] = G[b * 256 + 15 * 16 + tid];  // s[q] from ones column
  __syncthreads();
  int c = blockIdx.x * 256 + tid;
  float m = mu [b * NCH + c];
  float s = isd[b * NCH + c];
  const float* m2 = M2 + (size_t)(b * NCH + c) * QP;
  float t[Q];
#pragma unroll
  for (int r = 0; r < Q; ++r) t[r] = s * (m2[r] - m * ss[r]);
  float* ob = out + (size_t)b * (Q * NCH);
#pragma unroll
  for (int q = 0; q < Q; ++q) {
    float v = 0.f;
#pragma unroll
    for (int r = 0; r < Q; ++r) v += Gi[q * QP + r] * t[r];
    ob[q * NCH + c] = v;
  }
}

extern "C" void kernel_launch(void* const* d_in, const int* in_sizes, int n_in,
                              void* d_out, int out_size, void* d_ws, size_t ws_size,
                              hipStream_t stream) {
  const float* x = (const float*)d_in[0];   // (32,1024,32,32)
  const float* W = (const float*)d_in[1];   // (1025,12)
  float* out = (float*)d_out;               // (32,1,12288)
  float* ws  = (float*)d_ws;
  // ws layout (floats): ~6.6 MB total
  float* mu    = ws;                 // 32*1024
  float* isd   = ws + 32768;         // 32*1024
  float* Wp    = ws + 65536;         // 32*1024*16
  float* biasp = ws + 589824;        // 32*16
  float* H     = ws + 590336;        // 32*1024*16
  float* G     = ws + 1114624;       // 32*256
  float* Ginv  = ws + 1122816;       // 32*256
  float* M2    = ws + 1131008;       // 32*1024*16

  k_stats    <<<dim3(128, 32), 256, 0, stream>>>(x, mu, isd);
  k_prep_w   <<<dim3(8, 32),   128, 0, stream>>>(W, isd, Wp);
  k_prep_bias<<<32,            256, 0, stream>>>(W, mu, Wp, biasp);
  k_gemm_h   <<<dim3(16, 32),  128, 0, stream>>>(x, Wp, biasp, H);
  k_gram     <<<32,             32, 0, stream>>>(H, G);
  k_solve    <<<32,            384, 0, stream>>>(G, Ginv);
  k_gemm_m2  <<<dim3(16, 32),  128, 0, stream>>>(x, H, M2);
  k_finalize <<<dim3(4, 32),   256, 0, stream>>>(M2, Ginv, G, mu, isd, out);
}